// InvoiceGCN_37443524887039
// MI455X (gfx1250) — compile-verified
//
#include <hip/hip_runtime.h>

typedef float v2f __attribute__((ext_vector_type(2)));
typedef float v8f __attribute__((ext_vector_type(8)));

// -------------------------------------------------------------------------
// Build fused B matrix, Bmat[in_f][Npad]:
//   cols [0,16*out_f)  : nn_w reshaped, B[i][f*out_f+o] = nn_w[f][i*out_f+o]
//   cols [16of,17of)   : nn_b          (per-edge coefficient == 1)
//   cols [17of,18of)   : root          (node self term)
//   cols >= 18*out_f   : zero pad (Npad = round-up to 32)
// -------------------------------------------------------------------------
__global__ void build_bmat_kernel(const float* __restrict__ nn_w,
                                  const float* __restrict__ nn_b,
                                  const float* __restrict__ root,
                                  float* __restrict__ Bmat,
                                  int in_f, int out_f, int Npad) {
    int idx = blockIdx.x * blockDim.x + threadIdx.x;
    int total = in_f * Npad;
    if (idx >= total) return;
    int i = idx / Npad;
    int c = idx - i * Npad;
    float v = 0.0f;
    if (c < 16 * out_f) {
        int f = c / out_f;
        int o = c - f * out_f;
        v = nn_w[(size_t)f * (in_f * out_f) + i * out_f + o];
    } else if (c < 17 * out_f) {
        v = nn_b[(size_t)i * out_f + (c - 16 * out_f)];
    } else if (c < 18 * out_f) {
        v = root[(size_t)i * out_f + (c - 17 * out_f)];
    }
    Bmat[idx] = v;
}

// -------------------------------------------------------------------------
// z = X @ Bmat  with V_WMMA_F32_16X16X4_F32 (exact fp32, no downcast).
// One wave computes a 16(M) x 32(N) tile: two 16x16 accumulators share the
// A registers; K-loop steps by 4 (8 iterations for in_f = 32).
// A 16x4 layout (2 VGPRs/lane): lane = kHalf*16 + m,
//   a.x = A[m][2*kHalf], a.y = A[m][2*kHalf+1]   (ISA 7.12.2)
// B 4x16 layout mirrors the K split; C/D: vgpr j holds M = j + 8*kHalf.
// -------------------------------------------------------------------------
__global__ __launch_bounds__(128)
void gemm_z_wmma(const float* __restrict__ X,    // [N, in_f]
                 const float* __restrict__ B,    // [in_f, Npad]
                 float* __restrict__ Z,          // [N, Npad]
                 int Nn, int in_f, int Npad, int n_mtiles, int n_npairs) {
    int wave = blockIdx.x * (blockDim.x >> 5) + (threadIdx.x >> 5);
    int total = n_mtiles * n_npairs;
    if (wave >= total) return;          // wave-uniform: EXEC stays all-ones

    int np = wave % n_npairs;
    int mt = wave / n_npairs;
    int lane = threadIdx.x & 31;
    int m    = lane & 15;
    int half = lane >> 4;               // 0 or 1
    int kq   = half << 1;               // 0 or 2
    int m0   = mt << 4;
    int n0   = np << 5;

    int arow = m0 + m; if (arow >= Nn) arow = Nn - 1;   // safe clamp (no-op for N=50000)
    const float* xr = X + (size_t)arow * in_f + kq;

    v8f c0 = {0.f,0.f,0.f,0.f,0.f,0.f,0.f,0.f};
    v8f c1 = {0.f,0.f,0.f,0.f,0.f,0.f,0.f,0.f};

    for (int k = 0; k < in_f; k += 4) {
        v2f a;
        a.x = xr[k];
        a.y = xr[k + 1];
        const float* bp = B + (size_t)(k + kq) * Npad + n0 + m;
        v2f b0, b1;
        b0.x = bp[0];
        b0.y = bp[Npad];
        b1.x = bp[16];
        b1.y = bp[Npad + 16];
        c0 = __builtin_amdgcn_wmma_f32_16x16x4_f32(false, a, false, b0,
                                                   (short)0, c0, false, false);
        c1 = __builtin_amdgcn_wmma_f32_16x16x4_f32(false, a, false, b1,
                                                   (short)0, c1, false, false);
    }

    #pragma unroll
    for (int j = 0; j < 8; ++j) {
        int mr = m0 + j + (half << 3);
        if (mr < Nn) {
            size_t row = (size_t)mr * Npad;
            Z[row + n0 + m]      = c0[j];
            Z[row + n0 + 16 + m] = c1[j];
        }
    }
}

// -------------------------------------------------------------------------
// Degree counts (layer-invariant; rebuilt each call for determinism).
// -------------------------------------------------------------------------
__global__ void count_kernel(const int* __restrict__ dst,
                             float* __restrict__ cnt, int E) {
    int e = blockIdx.x * blockDim.x + threadIdx.x;
    if (e < E) atomicAdd(&cnt[dst[e]], 1.0f);
}

// -------------------------------------------------------------------------
// Per-edge combine: msg = z_bias + sum_f ea[f] * z[f], scatter-add to acc.
// One wave per edge; lane == output channel. z row gathers are 128B
// contiguous per f and L2-resident (z fits in the 192MB L2). ea loads are
// same-address across the wave -> broadcast.
// -------------------------------------------------------------------------
__global__ __launch_bounds__(256)
void edge_msg_kernel(const float* __restrict__ z,
                     const float* __restrict__ ea,
                     const int* __restrict__ src,
                     const int* __restrict__ dst,
                     float* __restrict__ acc,
                     int E, int out_f, int zstride) {
    int t = blockIdx.x * blockDim.x + threadIdx.x;
    int e = t >> 5;
    int lane = t & 31;
    if (e >= E) return;
    int s = src[e];
    int d = dst[e];
    const float* zr  = z + (size_t)s * zstride;
    const float* eae = ea + (size_t)e * 16;
    if (lane < out_f) {
        float msg = zr[16 * out_f + lane];      // nn_b channel (coeff 1)
        #pragma unroll
        for (int f = 0; f < 16; ++f)
            msg += eae[f] * zr[f * out_f + lane];
        atomicAdd(&acc[(size_t)d * out_f + lane], msg);
    }
}

// -------------------------------------------------------------------------
// out = [relu]( z_root + acc / max(cnt,1) + bias )
// -------------------------------------------------------------------------
__global__ void node_fin_kernel(const float* __restrict__ z,
                                const float* __restrict__ acc,
                                const float* __restrict__ cnt,
                                const float* __restrict__ bias,
                                float* __restrict__ out,
                                int Nn, int out_f, int zstride, int do_relu) {
    int idx = blockIdx.x * blockDim.x + threadIdx.x;
    if (idx >= Nn * out_f) return;
    int v = idx / out_f;
    int o = idx - v * out_f;
    float c = cnt[v];
    c = c > 1.0f ? c : 1.0f;
    float val = z[(size_t)v * zstride + 17 * out_f + o] + acc[idx] / c + bias[o];
    if (do_relu && val < 0.0f) val = 0.0f;
    out[idx] = val;
}

// -------------------------------------------------------------------------
extern "C" void kernel_launch(void* const* d_in, const int* in_sizes, int n_in,
                              void* d_out, int out_size, void* d_ws, size_t ws_size,
                              hipStream_t stream) {
    const float* x   = (const float*)d_in[0];
    const float* ea  = (const float*)d_in[1];
    const int*   ei  = (const int*)d_in[2];
    const int Nn = in_sizes[0] / 32;     // 50000
    const int E  = in_sizes[2] / 2;      // 400000
    const int* src = ei;
    const int* dst = ei + E;

    // workspace layout (all fp32): z | acc | cnt | h1 | h2 | Bmat  (~135 MB)
    const int ZMAX = 576;                 // 18 * HID_F, padded stride
    float* ws   = (float*)d_ws;
    float* z    = ws;
    float* acc  = z    + (size_t)Nn * ZMAX;
    float* cnt  = acc  + (size_t)Nn * 32;
    float* h1   = cnt  + Nn;
    float* h2   = h1   + (size_t)Nn * 32;
    float* Bmat = h2   + (size_t)Nn * 32;

    // degree counts once per call
    hipMemsetAsync(cnt, 0, (size_t)Nn * sizeof(float), stream);
    count_kernel<<<(E + 255) / 256, 256, 0, stream>>>(dst, cnt, E);

    const float* nw[4] = {(const float*)d_in[3],  (const float*)d_in[7],
                          (const float*)d_in[11], (const float*)d_in[15]};
    const float* nb[4] = {(const float*)d_in[4],  (const float*)d_in[8],
                          (const float*)d_in[12], (const float*)d_in[16]};
    const float* rt[4] = {(const float*)d_in[5],  (const float*)d_in[9],
                          (const float*)d_in[13], (const float*)d_in[17]};
    const float* bi[4] = {(const float*)d_in[6],  (const float*)d_in[10],
                          (const float*)d_in[14], (const float*)d_in[18]};
    const int inf_[4]  = {32, 32, 32, 32};
    const int outf[4]  = {32, 32, 32, 5};
    const float* hin[4] = {x, h1, h2, h1};
    float* hout[4]      = {h1, h2, h1, (float*)d_out};
    const int relu[4]   = {1, 1, 1, 0};

    for (int l = 0; l < 4; ++l) {
        const int in_f  = inf_[l];
        const int out_f = outf[l];
        const int Npad  = ((18 * out_f + 31) / 32) * 32;   // 576 or 96

        build_bmat_kernel<<<(in_f * Npad + 255) / 256, 256, 0, stream>>>(
            nw[l], nb[l], rt[l], Bmat, in_f, out_f, Npad);

        const int n_mt = (Nn + 15) / 16;     // 3125
        const int n_np = Npad / 32;          // 18 or 3
        const int total_waves = n_mt * n_np;
        gemm_z_wmma<<<(total_waves + 3) / 4, 128, 0, stream>>>(
            hin[l], Bmat, z, Nn, in_f, Npad, n_mt, n_np);

        hipMemsetAsync(acc, 0, (size_t)Nn * out_f * sizeof(float), stream);
        edge_msg_kernel<<<(E + 7) / 8, 256, 0, stream>>>(
            z, ea, src, dst, acc, E, out_f, Npad);

        node_fin_kernel<<<(Nn * out_f + 255) / 256, 256, 0, stream>>>(
            z, acc, cnt, bi[l], hout[l], Nn, out_f, Npad, relu[l]);
    }
}